// SpectralConv2d_37847251812691
// MI455X (gfx1250) — compile-verified
//
#include <hip/hip_runtime.h>

typedef float v2f __attribute__((ext_vector_type(2)));
typedef float v8f __attribute__((ext_vector_type(8)));

#define NIMG 4096  // B*CIN = COUT*CIN = 64*64

// workspace layout (in floats)
static constexpr size_t O_CAS    = 0;                      // 64x128  cas(+2*pi*p*m/128), p = rowsel(i)
static constexpr size_t O_CASM   = O_CAS    + 64 * 128;    // 64x128  cas(-...)
static constexpr size_t O_COS    = O_CASM   + 64 * 128;    // 32x128  cos(2*pi*q*n/128)
static constexpr size_t O_SIN    = O_COS    + 32 * 128;    // 32x128  sin
static constexpr size_t O_CAS32  = O_SIN    + 32 * 128;    // 32x32
static constexpr size_t O_CASM32 = O_CAS32  + 32 * 32;
static constexpr size_t O_COS32  = O_CASM32 + 32 * 32;
static constexpr size_t O_SIN32  = O_COS32  + 32 * 32;
static constexpr size_t O_XF     = O_SIN32  + 32 * 32;             // [4096][64][32]
static constexpr size_t O_WF     = O_XF + (size_t)NIMG * 64 * 32;  // [2][4096][32][32]
static constexpr size_t O_CF     = O_WF + (size_t)2 * NIMG * 1024; // [4096][64][32]

// ---------------- WMMA f32 16x16x4 helpers (wave32) ----------------
// A (16x4): lane L holds A[L%16][2*(L/16)+j] in vgpr j        (ISA 7.12.2)
// B (4x16): lane L holds B[2*(L/16)+j][L%16] in vgpr j
// C/D (16x16): lane L vgpr v holds C[v+8*(L/16)][L%16]
static __device__ inline v2f mk2(float a, float b) { v2f r; r.x = a; r.y = b; return r; }

static __device__ inline v2f load_a(const float* A, int lda, int m0, int k0, int lr, int hi) {
  const float* p = A + (size_t)(m0 + lr) * lda + (k0 + 2 * hi);
  return mk2(p[0], p[1]);
}
// A[m][k] = At[k][m]
static __device__ inline v2f load_at(const float* At, int ldt, int m0, int k0, int lr, int hi) {
  const float* p = At + (size_t)(k0 + 2 * hi) * ldt + (m0 + lr);
  return mk2(p[0], p[ldt]);
}
static __device__ inline v2f load_b(const float* B, int ldb, int k0, int n0, int lr, int hi) {
  const float* p = B + (size_t)(k0 + 2 * hi) * ldb + (n0 + lr);
  return mk2(p[0], p[ldb]);
}
// B[k][n] = Bt[n][k]
static __device__ inline v2f load_bt(const float* Bt, int ldt, int k0, int n0, int lr, int hi) {
  const float* p = Bt + (size_t)(n0 + lr) * ldt + (k0 + 2 * hi);
  return mk2(p[0], p[1]);
}
static __device__ inline v8f wmma4(v2f a, v2f b, v8f c) {
  return __builtin_amdgcn_wmma_f32_16x16x4_f32(false, a, false, b, (short)0, c, false, false);
}

// ---------------- K0: basis matrices ----------------
__global__ void k_basis(float* __restrict__ ws) {
  const int tid = threadIdx.x;
  const float w128 = 6.28318530717958647692f / 128.0f;
  const float w32  = 6.28318530717958647692f / 32.0f;
  for (int idx = tid; idx < 64 * 128; idx += blockDim.x) {
    int i = idx >> 7, m = idx & 127;
    int p = (i < 32) ? i : (i + 64);  // rows 0..31 then 96..127
    int r = (p * m) & 127;
    float s, c; sincosf(r * w128, &s, &c);
    ws[O_CAS  + idx] = c + s;
    ws[O_CASM + idx] = c - s;
  }
  for (int idx = tid; idx < 32 * 128; idx += blockDim.x) {
    int q = idx >> 7, n = idx & 127;
    int r = (q * n) & 127;
    float s, c; sincosf(r * w128, &s, &c);
    ws[O_COS + idx] = c;
    ws[O_SIN + idx] = s;
  }
  for (int idx = tid; idx < 32 * 32; idx += blockDim.x) {
    int p = idx >> 5, m = idx & 31;
    int r = (p * m) & 31;
    float s, c; sincosf(r * w32, &s, &c);
    ws[O_CAS32  + idx] = c + s;
    ws[O_CASM32 + idx] = c - s;
    ws[O_COS32  + idx] = c;
    ws[O_SIN32  + idx] = s;
  }
}

// ---------------- K1: partial 128-pt DHT of x -> xf[img][64][32] ----------------
// XF = (1/16384) * ( CAS * (X*COS^T)  +  CASM * (X*SIN^T) )
__global__ __launch_bounds__(256) void k_xdht(const float* __restrict__ x, float* __restrict__ ws) {
  __shared__ float Xs[32 * 132];   // 32-row chunk of X, padded
  __shared__ float Rs[128 * 68];   // R1 (cols 0..31) | R2 (cols 32..63)
  const int img  = blockIdx.x;
  const int tid  = threadIdx.x;
  const int wave = tid >> 5;
  const int lane = tid & 31;
  const int lr = lane & 15, hi = lane >> 4;
  const float* xg    = x + (size_t)img * (128 * 128);
  const float* bCOS  = ws + O_COS;
  const float* bSIN  = ws + O_SIN;
  const float* bCAS  = ws + O_CAS;
  const float* bCASM = ws + O_CASM;

  // Stage A: R1 = X*COS^T, R2 = X*SIN^T (128x32 each), processed in 32-row chunks
  for (int ch = 0; ch < 4; ++ch) {
    for (int idx = tid; idx < 32 * 128; idx += 256) {
      int rr = idx >> 7, cc = idx & 127;
      Xs[rr * 132 + cc] = xg[(size_t)(ch * 32 + rr) * 128 + cc];
    }
    __syncthreads();
    const int sel = wave >> 2, tm = (wave >> 1) & 1, tn = wave & 1;  // 8 waves = 8 tiles
    const float* Bt = sel ? bSIN : bCOS;  // B[k][q] = Bt[q][k]
    v8f acc = {};
    for (int k0 = 0; k0 < 128; k0 += 4) {
      v2f a = load_a(Xs, 132, tm * 16, k0, lr, hi);
      v2f b = load_bt(Bt, 128, k0, tn * 16, lr, hi);
      acc = wmma4(a, b, acc);
    }
    float* D = Rs + (size_t)(ch * 32 + tm * 16) * 68 + sel * 32 + tn * 16;
#pragma unroll
    for (int v = 0; v < 8; ++v) D[(v + 8 * hi) * 68 + lr] = acc[v];
    __syncthreads();
  }

  // Stage B: XF(64x32) = CAS*R1 + CASM*R2 ; one 16x16 tile per wave
  {
    const int tm = wave >> 1, tn = wave & 1;  // tm 0..3, tn 0..1
    v8f acc = {};
    for (int k0 = 0; k0 < 128; k0 += 4) {
      v2f a = load_a(bCAS, 128, tm * 16, k0, lr, hi);
      v2f b = load_b(Rs, 68, k0, tn * 16, lr, hi);
      acc = wmma4(a, b, acc);
    }
    for (int k0 = 0; k0 < 128; k0 += 4) {
      v2f a = load_a(bCASM, 128, tm * 16, k0, lr, hi);
      v2f b = load_b(Rs + 32, 68, k0, tn * 16, lr, hi);
      acc = wmma4(a, b, acc);
    }
    const float sc = 1.0f / 16384.0f;  // forward-norm DHT over 128x128
    float* D = ws + O_XF + (size_t)img * 2048 + (size_t)(tm * 16) * 32 + tn * 16;
#pragma unroll
    for (int v = 0; v < 8; ++v) D[(v + 8 * hi) * 32 + lr] = acc[v] * sc;
  }
}

// ---------------- 32-pt DHT core (shared by K2/K3): Ts = [CAS32;CASM32]*Src ----------------
static __device__ inline void dht32_stageA(const float* __restrict__ ws, const float* Src,
                                           float* Ts, int wave, int lr, int hi) {
  for (int t = wave * 2; t < wave * 2 + 2; ++t) {   // 8 tiles over 4 waves
    const int tm = t >> 1, tn = t & 1;              // tm 0..3
    const float* A = (tm < 2) ? (ws + O_CAS32) : (ws + O_CASM32);
    const int am0 = (tm & 1) * 16;
    v8f acc = {};
    for (int k0 = 0; k0 < 32; k0 += 4) {
      v2f a = load_a(A, 32, am0, k0, lr, hi);
      v2f b = load_b(Src, 36, k0, tn * 16, lr, hi);
      acc = wmma4(a, b, acc);
    }
    float* D = Ts + (size_t)(tm * 16) * 36 + tn * 16;
#pragma unroll
    for (int v = 0; v < 8; ++v) D[(v + 8 * hi) * 36 + lr] = acc[v];
  }
}
// acc = T*COS32^T + Tm*SIN32^T for tile (tm,tn)
static __device__ inline v8f dht32_stageB(const float* __restrict__ ws, const float* Ts,
                                          int tm, int tn, int lr, int hi) {
  v8f acc = {};
  for (int k0 = 0; k0 < 32; k0 += 4) {
    v2f a = load_a(Ts, 36, tm * 16, k0, lr, hi);
    v2f b = load_bt(ws + O_COS32, 32, k0, tn * 16, lr, hi);
    acc = wmma4(a, b, acc);
  }
  for (int k0 = 0; k0 < 32; k0 += 4) {
    v2f a = load_a(Ts + 32 * 36, 36, tm * 16, k0, lr, hi);
    v2f b = load_bt(ws + O_SIN32, 32, k0, tn * 16, lr, hi);
    acc = wmma4(a, b, acc);
  }
  return acc;
}

// ---------------- K2: 32-pt DHT of weights -> wf[2][4096][32][32] ----------------
__global__ __launch_bounds__(128) void k_wdht(const float* __restrict__ w1,
                                              const float* __restrict__ w2,
                                              float* __restrict__ ws) {
  __shared__ float Ws[32 * 36];
  __shared__ float Ts[64 * 36];
  const int id = blockIdx.x;           // 0..8191
  const int half = id >> 12;
  const int imgw = id & 4095;
  const int tid = threadIdx.x, wave = tid >> 5, lane = tid & 31;
  const int lr = lane & 15, hi = lane >> 4;
  const float* src = (half ? w2 : w1) + (size_t)imgw * 1024;
  for (int idx = tid; idx < 1024; idx += 128) Ws[(idx >> 5) * 36 + (idx & 31)] = src[idx];
  __syncthreads();
  dht32_stageA(ws, Ws, Ts, wave, lr, hi);
  __syncthreads();
  {
    const int tm = wave >> 1, tn = wave & 1;
    v8f acc = dht32_stageB(ws, Ts, tm, tn, lr, hi);
    const float sc = 1.0f / 1024.0f;   // forward-norm 32x32 DHT
    float* D = ws + O_WF + (size_t)id * 1024 + (size_t)(tm * 16) * 32 + tn * 16;
#pragma unroll
    for (int v = 0; v < 8; ++v) D[(v + 8 * hi) * 32 + lr] = acc[v] * sc;
  }
}

// ---------------- K3: c = (1/64)*DHT32( DHT32(xf_half) .* wf ) -> cf ----------------
__global__ __launch_bounds__(128) void k_spec(float* __restrict__ ws) {
  __shared__ float Zs[32 * 36];
  __shared__ float Ts[64 * 36];
  __shared__ float Ps[32 * 36];
  const int id = blockIdx.x;           // img*2 + half
  const int img = id >> 1, half = id & 1;
  const int tid = threadIdx.x, wave = tid >> 5, lane = tid & 31;
  const int lr = lane & 15, hi = lane >> 4;
  const float* xf = ws + O_XF + (size_t)img * 2048 + (size_t)half * 1024;
  for (int idx = tid; idx < 1024; idx += 128) Zs[(idx >> 5) * 36 + (idx & 31)] = xf[idx];
  __syncthreads();
  dht32_stageA(ws, Zs, Ts, wave, lr, hi);     // T of xf slice
  __syncthreads();
  {  // G = (1/1024)*DHT32(xf) ; P = G .* WF
    const int tm = wave >> 1, tn = wave & 1;
    v8f acc = dht32_stageB(ws, Ts, tm, tn, lr, hi);
    const float* wf = ws + O_WF + ((size_t)half * NIMG + img) * 1024 + (size_t)(tm * 16) * 32 + tn * 16;
    float* D = Ps + (size_t)(tm * 16) * 36 + tn * 16;
    const float g1 = 1.0f / 1024.0f;
#pragma unroll
    for (int v = 0; v < 8; ++v) {
      int rr = v + 8 * hi;
      D[rr * 36 + lr] = (acc[v] * g1) * wf[rr * 32 + lr];
    }
  }
  __syncthreads();
  dht32_stageA(ws, Ps, Ts, wave, lr, hi);     // T of product
  __syncthreads();
  {  // c = (1/64)*(1/1024)*unnormalized DHT32
    const int tm = wave >> 1, tn = wave & 1;
    v8f acc = dht32_stageB(ws, Ts, tm, tn, lr, hi);
    const float sc = 1.0f / 65536.0f;
    float* D = ws + O_CF + (size_t)img * 2048 + (size_t)half * 1024 + (size_t)(tm * 16) * 32 + tn * 16;
#pragma unroll
    for (int v = 0; v < 8; ++v) D[(v + 8 * hi) * 32 + lr] = acc[v] * sc;
  }
}

// ---------------- K4: out = (1/(64*16384)) * ( CAS^T*cf*COS + CASM^T*cf*SIN ) ----------------
__global__ __launch_bounds__(256) void k_out(float* __restrict__ ws, float* __restrict__ out) {
  __shared__ float Cs[64 * 36];
  __shared__ float Ts[256 * 36];   // T rows 0..127, Tm rows 128..255
  const int img = blockIdx.x;
  const int tid = threadIdx.x, wave = tid >> 5, lane = tid & 31;
  const int lr = lane & 15, hi = lane >> 4;
  const float* cf = ws + O_CF + (size_t)img * 2048;
  for (int idx = tid; idx < 2048; idx += 256) Cs[(idx >> 5) * 36 + (idx & 31)] = cf[idx];
  __syncthreads();

  // Stage A: T[m][q] = sum_i CAS[i][m]*cf[i][q] (and Tm with CASM); K=64
  for (int sel = 0; sel < 2; ++sel) {
    const float* At = sel ? (ws + O_CASM) : (ws + O_CAS);  // A[m][k] = At[k][m]
    for (int tn = 0; tn < 2; ++tn) {
      v8f acc = {};
      for (int k0 = 0; k0 < 64; k0 += 4) {
        v2f a = load_at(At, 128, wave * 16, k0, lr, hi);
        v2f b = load_b(Cs, 36, k0, tn * 16, lr, hi);
        acc = wmma4(a, b, acc);
      }
      float* D = Ts + (size_t)(sel * 128 + wave * 16) * 36 + tn * 16;
#pragma unroll
      for (int v = 0; v < 8; ++v) D[(v + 8 * hi) * 36 + lr] = acc[v];
    }
  }
  __syncthreads();

  // Stage B: out(128x128) = T*COS + Tm*SIN ; wave owns column block tn=wave
  const float sc = 1.0f / (16384.0f * 64.0f);
  for (int tm = 0; tm < 8; ++tm) {
    v8f acc = {};
    for (int k0 = 0; k0 < 32; k0 += 4) {
      v2f a = load_a(Ts, 36, tm * 16, k0, lr, hi);
      v2f b = load_b(ws + O_COS, 128, k0, wave * 16, lr, hi);
      acc = wmma4(a, b, acc);
    }
    for (int k0 = 0; k0 < 32; k0 += 4) {
      v2f a = load_a(Ts + 128 * 36, 36, tm * 16, k0, lr, hi);
      v2f b = load_b(ws + O_SIN, 128, k0, wave * 16, lr, hi);
      acc = wmma4(a, b, acc);
    }
    float* D = out + (size_t)img * 16384 + (size_t)(tm * 16) * 128 + wave * 16;
#pragma unroll
    for (int v = 0; v < 8; ++v) D[(v + 8 * hi) * 128 + lr] = acc[v] * sc;
  }
}

extern "C" void kernel_launch(void* const* d_in, const int* in_sizes, int n_in,
                              void* d_out, int out_size, void* d_ws, size_t ws_size,
                              hipStream_t stream) {
  (void)in_sizes; (void)n_in; (void)out_size; (void)ws_size;
  const float* x  = (const float*)d_in[0];
  const float* w1 = (const float*)d_in[1];
  const float* w2 = (const float*)d_in[2];
  float* out = (float*)d_out;
  float* ws  = (float*)d_ws;

  k_basis<<<1, 256, 0, stream>>>(ws);
  k_wdht <<<2 * NIMG, 128, 0, stream>>>(w1, w2, ws);
  k_xdht <<<NIMG, 256, 0, stream>>>(x, ws);
  k_spec <<<2 * NIMG, 128, 0, stream>>>(ws);
  k_out  <<<NIMG, 256, 0, stream>>>(ws, out);
}